// GINConv_22402549415971
// MI455X (gfx1250) — compile-verified
//
#include <hip/hip_runtime.h>

typedef float v2f __attribute__((ext_vector_type(2)));
typedef float v8f __attribute__((ext_vector_type(8)));

#define D_FEAT 128
#define ZS 132  // padded LDS row stride (dwords): 132*4 % 256 = 16 -> conflict-free strided b64 reads

// ---------------------------------------------------------------------------
// Kernel 1: edge scatter-max.  One wave (32 lanes) per edge; each lane moves
// one float4 (4 features) and issues 4 native u32 atomic-max ops on the
// monotonic float->uint key.  Aggregate key array lives in d_ws (L2-resident).
// ---------------------------------------------------------------------------
__device__ __forceinline__ unsigned f2key(float x) {
    unsigned b = __float_as_uint(x);
    return (b & 0x80000000u) ? ~b : (b | 0x80000000u);
}

__global__ __launch_bounds__(256)
void gin_scatter_max(const float4* __restrict__ h4,
                     const int* __restrict__ src,
                     const int* __restrict__ dst,
                     unsigned* __restrict__ aggkey,
                     int n_edges)
{
    const int t    = blockIdx.x * blockDim.x + threadIdx.x;
    const int edge = t >> 5;
    const int lane = t & 31;
    if (edge >= n_edges) return;

    const int s = src[edge];
    const int d = dst[edge];

    // h row = 32 float4's; lane grabs its float4 (global_load_b128)
    float4 v = h4[(size_t)s * 32 + lane];

    unsigned* kp = aggkey + (size_t)d * D_FEAT + lane * 4;
    atomicMax(kp + 0, f2key(v.x));
    atomicMax(kp + 1, f2key(v.y));
    atomicMax(kp + 2, f2key(v.z));
    atomicMax(kp + 3, f2key(v.w));
}

// ---------------------------------------------------------------------------
// Kernel 2: fused  z = h + decode(aggkey);  hdn = relu(z@W1+b1);
//                  out = hdn@W2 + b2         via V_WMMA_F32_16X16X4_F32.
// One workgroup = 16 rows (exactly 50000/16 = 3125 blocks), 8 waves,
// wave w owns output columns [16w, 16w+16).
//
// A-fragment layout (16x4 f32): lane l holds M = l%16, K = k + 2*(l/16) .. +1
//   -> one ds_load_b64 from the padded LDS tile per k-step.
// B-fragment (4x16): lane l holds column c0 + l%16, rows K = k + 2*(l/16)..+1
//   -> two global_load_b32 (L2-hot: W1+W2 = 128 KB total).
// C/D (16x16 f32, 8 VGPRs): VGPR i -> M = i + 8*(l/16), N = l%16.
// ---------------------------------------------------------------------------
__global__ __launch_bounds__(256)
void gin_mlp(const float* __restrict__ h,
             const unsigned* __restrict__ aggkey,
             const float* __restrict__ W1, const float* __restrict__ b1,
             const float* __restrict__ W2, const float* __restrict__ b2,
             float* __restrict__ out)
{
    __shared__ float zs[16 * ZS];
    __shared__ float hs[16 * ZS];

    const int tid  = threadIdx.x;
    const int wave = tid >> 5;
    const int lane = tid & 31;
    const int row0 = blockIdx.x * 16;

    // ---- stage z = h + decode(max-key); key==0 => isolated node => agg 0 ----
    for (int i = tid; i < 16 * D_FEAT; i += 256) {
        const int r = i >> 7;
        const int f = i & 127;
        const unsigned k = aggkey[(size_t)(row0 + r) * D_FEAT + f];
        float a = 0.0f;
        if (k != 0u) {
            const unsigned bits = (k & 0x80000000u) ? (k ^ 0x80000000u) : ~k;
            a = __uint_as_float(bits);
        }
        zs[r * ZS + f] = h[(size_t)(row0 + r) * D_FEAT + f] + a;
    }
    __syncthreads();

    const int c0 = wave * 16;   // this wave's output-column block
    const int nl = lane & 15;   // N index (and A-row M index)
    const int hi = lane >> 4;   // K-pair selector

    // ---------------- layer 1: hdn = relu(z @ W1 + b1) ----------------
    v8f acc = {};
    {
        const float* arow = &zs[nl * ZS + hi * 2];
        const float* bcol = W1 + (size_t)(hi * 2) * D_FEAT + c0 + nl;
        #pragma unroll
        for (int k = 0; k < D_FEAT; k += 4) {
            v2f a = *(const v2f*)(arow + k);
            v2f b;
            b.x = bcol[(size_t)k * D_FEAT];
            b.y = bcol[(size_t)(k + 1) * D_FEAT];
            acc = __builtin_amdgcn_wmma_f32_16x16x4_f32(
                false, a, false, b, (short)0, acc, false, false);
        }
    }
    {
        const float bb = b1[c0 + nl];
        #pragma unroll
        for (int i = 0; i < 8; ++i) {
            float v = acc[i] + bb;
            v = v > 0.0f ? v : 0.0f;                 // ReLU
            hs[(i + hi * 8) * ZS + c0 + nl] = v;     // M = i + 8*hi, N = c0+nl
        }
    }
    __syncthreads();

    // ---------------- layer 2: out = hdn @ W2 + b2 ----------------
    v8f acc2 = {};
    {
        const float* arow = &hs[nl * ZS + hi * 2];
        const float* bcol = W2 + (size_t)(hi * 2) * D_FEAT + c0 + nl;
        #pragma unroll
        for (int k = 0; k < D_FEAT; k += 4) {
            v2f a = *(const v2f*)(arow + k);
            v2f b;
            b.x = bcol[(size_t)k * D_FEAT];
            b.y = bcol[(size_t)(k + 1) * D_FEAT];
            acc2 = __builtin_amdgcn_wmma_f32_16x16x4_f32(
                false, a, false, b, (short)0, acc2, false, false);
        }
    }
    {
        const float bb = b2[c0 + nl];
        #pragma unroll
        for (int i = 0; i < 8; ++i) {
            const int m = i + hi * 8;
            out[(size_t)(row0 + m) * D_FEAT + c0 + nl] = acc2[i] + bb;
        }
    }
}

// ---------------------------------------------------------------------------
// Host launcher (graph-capture safe: only async stream ops)
// ---------------------------------------------------------------------------
extern "C" void kernel_launch(void* const* d_in, const int* in_sizes, int n_in,
                              void* d_out, int out_size, void* d_ws, size_t ws_size,
                              hipStream_t stream) {
    const float* h   = (const float*)d_in[0];
    const int*   src = (const int*)d_in[1];
    const int*   dst = (const int*)d_in[2];
    const float* W1  = (const float*)d_in[3];
    const float* b1  = (const float*)d_in[4];
    const float* W2  = (const float*)d_in[5];
    const float* b2  = (const float*)d_in[6];
    float* out = (float*)d_out;

    const int n_nodes = in_sizes[0] / D_FEAT;   // 50000
    const int n_edges = in_sizes[1];            // 800000

    unsigned* aggkey = (unsigned*)d_ws;         // n_nodes*128*4 = 25.6 MB
    const size_t key_bytes = (size_t)n_nodes * D_FEAT * sizeof(unsigned);

    // 1) init max-keys to 0 (== "no edge" sentinel, below every finite key)
    hipMemsetAsync(aggkey, 0, key_bytes, stream);

    // 2) scatter-max: one wave per edge
    {
        const long long threads = (long long)n_edges * 32;
        const int blocks = (int)((threads + 255) / 256);
        gin_scatter_max<<<blocks, 256, 0, stream>>>((const float4*)h, src, dst,
                                                    aggkey, n_edges);
    }

    // 3) fused decode + 2-layer MLP with f32 WMMA; 16 rows per block
    {
        const int blocks = n_nodes / 16;        // 3125, exact
        gin_mlp<<<blocks, 256, 0, stream>>>(h, aggkey, W1, b1, W2, b2, out);
    }
    (void)n_in; (void)out_size; (void)ws_size;
}